// FrequencyLayer_74878459838534
// MI455X (gfx1250) — compile-verified
//
#include <hip/hip_runtime.h>
#include <stdint.h>

#if defined(__AMDGCN__) && __has_builtin(__builtin_amdgcn_tensor_load_to_lds)
#define USE_TDM 1
#else
#define USE_TDM 0
#endif

typedef __attribute__((ext_vector_type(4))) unsigned int u32x4;
typedef __attribute__((ext_vector_type(4))) int i32x4;
typedef __attribute__((ext_vector_type(8))) int i32x8;

#if USE_TDM
// Build a 2D-tile Tensor DMA descriptor (D#) per CDNA5 ISA
// §8.3/§8.4 and issue TENSOR_LOAD_TO_LDS. Tile: nrows x row_elems f32, rows
// contiguous in global memory with stride row_elems.
__device__ __forceinline__ void tdm_load_rows(const float* gsrc, uint32_t lds_byte,
                                              uint32_t nrows, uint32_t row_elems) {
  uint64_t ga = (uint64_t)(uintptr_t)gsrc;
  u32x4 g0;
  g0[0] = 1u;                                            // count=1 (valid), user mode
  g0[1] = lds_byte;                                      // lds_addr (bytes)
  g0[2] = (uint32_t)ga;                                  // global_addr[31:0]
  g0[3] = (uint32_t)((ga >> 32) & 0x01FFFFFFu) | (2u << 30); // addr[56:32] | type=2
  i32x8 g1;
  g1[0] = (int)(2u << 16);               // workgroup_mask=0, data_size=2 (4B), flags=0
  g1[1] = (int)(row_elems << 16);        // tensor_dim0[15:0] in [31:16]
  g1[2] = (int)(nrows << 16);            // tensor_dim0 hi=0 | tensor_dim1[15:0]
  g1[3] = (int)(row_elems << 16);        // tensor_dim1 hi=0 | tile_dim0
  g1[4] = (int)nrows;                    // tile_dim1 | tile_dim2=0
  g1[5] = (int)row_elems;                // tensor_dim0_stride[31:0]
  g1[6] = 0;                             // stride hi | tensor_dim1_stride lo
  g1[7] = 0;
  i32x4 z4 = {0, 0, 0, 0};
#if __clang_major__ >= 23
  i32x8 z8 = {0, 0, 0, 0, 0, 0, 0, 0};
  __builtin_amdgcn_tensor_load_to_lds(g0, g1, z4, z4, z8, 0);
#else
  __builtin_amdgcn_tensor_load_to_lds(g0, g1, z4, z4, 0);
#endif
}
#endif

// B=64, S=512, H=1024. Each block: one s-octet (8 consecutive s) x all H,
// for NB=4 batches. 256 threads, thread owns 4 contiguous channels (float4).
__global__ __launch_bounds__(256) void freqlayer_kernel(
    const float* __restrict__ xin, const float* __restrict__ ww0,
    const float* __restrict__ ww1, const float* __restrict__ ww2,
    const float* __restrict__ srt, const float* __restrict__ sqb,
    const float* __restrict__ lnw, const float* __restrict__ lnb,
    float* __restrict__ out) {
  constexpr int H = 1024;
  constexpr int S = 512;
  constexpr int NB = 4;
  __shared__ float sm[11 * H];
  __shared__ float red[16];

  const int tid = (int)threadIdx.x;
  const int g   = (int)blockIdx.x;        // 0..63 : s-octet index
  const int b0  = (int)blockIdx.y * NB;   // batch base

  float* s_ww0 = sm + 0 * H;  // 4 rows: cd1 positions 4g..4g+3
  float* s_ww1 = sm + 4 * H;  // 2 rows: cd2 positions 2g..2g+1
  float* s_ww2 = sm + 6 * H;  // 1 row : cd3 position g
  float* s_srt = sm + 7 * H;
  float* s_sqb = sm + 8 * H;
  float* s_lnw = sm + 9 * H;
  float* s_lnb = sm + 10 * H;

#if USE_TDM
  if (tid < 32) {  // wave 0 drives the Tensor Data Mover
    uint32_t base = (uint32_t)(uintptr_t)sm;  // low 32 bits of LDS flat addr = LDS offset
    tdm_load_rows(ww0 + (size_t)(4 * g) * H, base + 0 * H * 4, 4, H);
    tdm_load_rows(ww1 + (size_t)(2 * g) * H, base + 4 * H * 4, 2, H);
    tdm_load_rows(ww2 + (size_t)g * H,       base + 6 * H * 4, 1, H);
    tdm_load_rows(srt,                        base + 7 * H * 4, 1, H);
    tdm_load_rows(sqb,                        base + 8 * H * 4, 1, H);
    tdm_load_rows(lnw,                        base + 9 * H * 4, 1, H);
    tdm_load_rows(lnb,                        base + 10 * H * 4, 1, H);
    __builtin_amdgcn_s_wait_tensorcnt(0);
  }
#else
  for (int i = tid; i < H; i += 256) {
#pragma unroll
    for (int k = 0; k < 4; ++k) s_ww0[k * H + i] = ww0[(size_t)(4 * g + k) * H + i];
    s_ww1[0 * H + i] = ww1[(size_t)(2 * g + 0) * H + i];
    s_ww1[1 * H + i] = ww1[(size_t)(2 * g + 1) * H + i];
    s_ww2[i] = ww2[(size_t)g * H + i];
    s_srt[i] = srt[i];
    s_sqb[i] = sqb[i];
    s_lnw[i] = lnw[i];
    s_lnb[i] = lnb[i];
  }
#endif
  __syncthreads();

  const int h0 = tid * 4;

  // Fold 1/sqrt(2) powers: low = (A3 ± w2*D3)/8 + (w1*D2)/4 + (w0*D1)/2
  float W0r[4][4], W1r[4][2], W2r[4], C1[4], C2[4], LW[4], LB[4];
#pragma unroll
  for (int c = 0; c < 4; ++c) {
    const int h = h0 + c;
    const float sr = s_srt[h];
    const float s2 = sr * sr;
#pragma unroll
    for (int k = 0; k < 4; ++k) W0r[c][k] = s_ww0[k * H + h] * s2 * 0.5f;
#pragma unroll
    for (int k = 0; k < 2; ++k) W1r[c][k] = s_ww1[k * H + h] * s2 * 0.25f;
    W2r[c] = s_ww2[h] * s2 * 0.125f;
    const float sb = s_sqb[h];
    const float sb2 = sb * sb;
    C1[c] = 1.0f - sb2;   // x = (1-sb^2)*low + (1+sb^2)*in
    C2[c] = 1.0f + sb2;
    LW[c] = s_lnw[h];
    LB[c] = s_lnb[h];
  }

  for (int bi = 0; bi < NB; ++bi) {
    const int b = b0 + bi;
    const float* src = xin + ((size_t)b * S + (size_t)(8 * g)) * H + h0;
    float*       dst = out + ((size_t)b * S + (size_t)(8 * g)) * H + h0;

    float4 v[8];
#pragma unroll
    for (int s = 0; s < 8; ++s) v[s] = *(const float4*)(src + (size_t)s * H);

    __syncthreads();                 // previous iteration's red[] reads done
    if (tid < 16) red[tid] = 0.0f;

    float x[8][4];
    float ps[8], pq[8];
#pragma unroll
    for (int s = 0; s < 8; ++s) { ps[s] = 0.0f; pq[s] = 0.0f; }

#pragma unroll
    for (int c = 0; c < 4; ++c) {
      float e[8];
#pragma unroll
      for (int s = 0; s < 8; ++s) e[s] = ((const float*)&v[s])[c];
      // unnormalized 3-level Haar on the octet
      const float A1_0 = e[0] + e[1], D1_0 = e[0] - e[1];
      const float A1_1 = e[2] + e[3], D1_1 = e[2] - e[3];
      const float A1_2 = e[4] + e[5], D1_2 = e[4] - e[5];
      const float A1_3 = e[6] + e[7], D1_3 = e[6] - e[7];
      const float A2_0 = A1_0 + A1_1, D2_0 = A1_0 - A1_1;
      const float A2_1 = A1_2 + A1_3, D2_1 = A1_2 - A1_3;
      const float A3 = A2_0 + A2_1, D3 = A2_0 - A2_1;
      // scaled inverse (normalization folded into W*r)
      const float u0 = A3 * 0.125f;
      const float w3 = W2r[c] * D3;
      const float p0 = u0 + w3, p1 = u0 - w3;
      const float q0 = W1r[c][0] * D2_0, q1 = W1r[c][1] * D2_1;
      const float n0 = p0 + q0, n1 = p0 - q0, n2 = p1 + q1, n3 = p1 - q1;
      const float r0 = W0r[c][0] * D1_0, r1 = W0r[c][1] * D1_1;
      const float r2 = W0r[c][2] * D1_2, r3 = W0r[c][3] * D1_3;
      float low[8];
      low[0] = n0 + r0; low[1] = n0 - r0;
      low[2] = n1 + r1; low[3] = n1 - r1;
      low[4] = n2 + r2; low[5] = n2 - r2;
      low[6] = n3 + r3; low[7] = n3 - r3;
#pragma unroll
      for (int s = 0; s < 8; ++s) {
        const float xv = C1[c] * low[s] + C2[c] * e[s];
        x[s][c] = xv;
        ps[s] += xv;
        pq[s] = fmaf(xv, xv, pq[s]);
      }
    }

    // wave32 reduction of per-s sums, then cross-wave via LDS float atomics
#pragma unroll
    for (int m = 16; m > 0; m >>= 1) {
#pragma unroll
      for (int s = 0; s < 8; ++s) {
        ps[s] += __shfl_xor(ps[s], m, 32);
        pq[s] += __shfl_xor(pq[s], m, 32);
      }
    }
    __syncthreads();                 // red[] zeroing visible
    if ((tid & 31) == 0) {
#pragma unroll
      for (int s = 0; s < 8; ++s) {
        atomicAdd(&red[s], ps[s]);       // ds_add_f32
        atomicAdd(&red[8 + s], pq[s]);
      }
    }
    __syncthreads();

#pragma unroll
    for (int s = 0; s < 8; ++s) {
      const float mean = red[s] * (1.0f / (float)H);
      const float var  = fmaf(-mean, mean, red[8 + s] * (1.0f / (float)H));
      const float rstd = rsqrtf(var + 1e-12f);
      float4 o;
#pragma unroll
      for (int c = 0; c < 4; ++c)
        ((float*)&o)[c] = fmaf((x[s][c] - mean) * rstd, LW[c], LB[c]);
      *(float4*)(dst + (size_t)s * H) = o;
    }
  }
}

extern "C" void kernel_launch(void* const* d_in, const int* in_sizes, int n_in,
                              void* d_out, int out_size, void* d_ws, size_t ws_size,
                              hipStream_t stream) {
  const float* xin = (const float*)d_in[0];
  const float* ww0 = (const float*)d_in[1];
  const float* ww1 = (const float*)d_in[2];
  const float* ww2 = (const float*)d_in[3];
  const float* srt = (const float*)d_in[4];
  const float* sqb = (const float*)d_in[5];
  const float* lnw = (const float*)d_in[6];
  const float* lnb = (const float*)d_in[7];
  float* out = (float*)d_out;
  dim3 grid(64, 16);  // 64 s-octets x 16 batch-quads (B=64, NB=4)
  freqlayer_kernel<<<grid, dim3(256), 0, stream>>>(xin, ww0, ww1, ww2, srt, sqb,
                                                   lnw, lnb, out);
  (void)in_sizes; (void)n_in; (void)out_size; (void)d_ws; (void)ws_size;
}